// UncertaintyEvalWrapper_6004364280779
// MI455X (gfx1250) — compile-verified
//
#include <hip/hip_runtime.h>
#include <cstdint>

// ---------------- problem constants ----------------
#define Bn   2
#define Hn   96
#define Wn   96
#define VNv  9
#define TNp  (Hn*Wn)        // 9216 pixels
#define RH   512
#define DH   4096
#define NTILE (TNp/16)      // 576 pixel tiles of 16
#define CHT  96             // pixel tiles staged in LDS per chunk (48KB per buffer)
#define NCHUNK (NTILE/CHT)  // 6
#define THR2 0.9801f        // 0.99^2
#define EPSF 1e-6f
#define BIGQ 1e30f          // masked-pixel scale -> vote always fails

typedef float v2f __attribute__((ext_vector_type(2)));
typedef float v8f __attribute__((ext_vector_type(8)));

// ---------------- workspace layout (floats) ----------------
#define OFF_PACKED 0
#define SZ_PACKED  (Bn*VNv*NTILE*32*4)   // wmma-ready B tiles (per-lane float4)
#define OFF_P      (OFF_PACKED + SZ_PACKED)
#define SZ_P       (Bn*VNv*TNp*4)        // plain table (ux,uy,c,uq)
#define OFF_FG     (OFF_P + SZ_P)
#define SZ_FG      (Bn)
#define OFF_MEAN   (OFF_FG + SZ_FG)
#define SZ_MEAN    (Bn*VNv*2)
#define OFF_CNTR   (OFF_MEAN + SZ_MEAN)
#define SZ_CNTR    (Bn*VNv*RH)
#define OFF_HYPR   (OFF_CNTR + SZ_CNTR)
#define SZ_HYPR    (Bn*VNv*RH*2)
#define OFF_CNTD   (OFF_HYPR + SZ_HYPR)
#define SZ_CNTD    (Bn*VNv*DH)
#define OFF_HYPD   (OFF_CNTD + SZ_CNTD)
#define SZ_HYPD    (Bn*VNv*DH*2)

// ---------------- small reductions (wave32) ----------------
__device__ __forceinline__ float wave_sum32(float v) {
#pragma unroll
  for (int m = 1; m < 32; m <<= 1) v += __shfl_xor(v, m, 32);
  return v;
}
__device__ __forceinline__ float block_sum(float v, float* s8) {
  v = wave_sum32(v);
  __syncthreads();
  if ((threadIdx.x & 31) == 0) s8[threadIdx.x >> 5] = v;
  __syncthreads();
  float r = 0.f;
#pragma unroll
  for (int i = 0; i < 8; ++i) r += s8[i];
  return r;
}
__device__ __forceinline__ float block_max(float v, float* s8) {
#pragma unroll
  for (int m = 1; m < 32; m <<= 1) v = fmaxf(v, __shfl_xor(v, m, 32));
  __syncthreads();
  if ((threadIdx.x & 31) == 0) s8[threadIdx.x >> 5] = v;
  __syncthreads();
  float r = -3.4e38f;
#pragma unroll
  for (int i = 0; i < 8; ++i) r = fmaxf(r, s8[i]);
  return r;
}

// ---------------- CDNA5 async global->LDS staging ----------------
__device__ __forceinline__ void async_ld_b128(uint32_t lds_byte_off, const void* gptr) {
  asm volatile("global_load_async_to_lds_b128 %0, %1, off"
               :: "v"(lds_byte_off), "v"(gptr) : "memory");
}
// wait until ALL of this wave's async loads have landed
__device__ __forceinline__ void wait_async0() {
  asm volatile("s_wait_asynccnt 0" ::: "memory");
}
// ASYNCcnt decrements in order: waiting <=12 means "everything except the
// 12 most recently issued loads (the prefetched next chunk) is complete".
__device__ __forceinline__ void wait_async_keep12() {
  asm volatile("s_wait_asynccnt 12" ::: "memory");
}

// ---------------- hypothesis generation (matches _gen_hyp) ----------------
__device__ __forceinline__ float2 gen_hyp(const float* __restrict__ vp,
                                          const int* __restrict__ idx2,
                                          int b, int v) {
  int t0 = idx2[0], t1 = idx2[1];
  const float* vx = vp + (size_t)(b*2*VNv + 2*v)*TNp;
  const float* vy = vx + TNp;
  float d0x = vx[t0], d0y = vy[t0];
  float d1x = vx[t1], d1y = vy[t1];
  float p0x = (float)(t0 % Wn), p0y = (float)(t0 / Wn);
  float p1x = (float)(t1 % Wn), p1y = (float)(t1 / Wn);
  float n0x = d0y, n0y = -d0x;
  float n1x = d1y, n1y = -d1x;
  float c0 = n0x*p0x + n0y*p0y;
  float c1 = n1x*p1x + n1y*p1y;
  float det = n0x*n1y - n0y*n1x;
  bool safe = fabsf(det) > EPSF;
  float ds = safe ? det : 1.f;
  float x = (c0*n1y - c1*n0y) / ds;
  float y = (n0x*c1 - n1x*c0) / ds;
  return safe ? make_float2(x, y) : make_float2(0.f, 0.f);
}

// ---------------- kernel 0: foreground count ----------------
__global__ __launch_bounds__(256) void fg_kernel(const int* __restrict__ mask,
                                                 float* __restrict__ fgw) {
  __shared__ float s8[8];
  int b = blockIdx.x;
  float s = 0.f;
  for (int t = threadIdx.x; t < TNp; t += 256) s += (float)mask[b*TNp + t];
  s = block_sum(s, s8);
  if (threadIdx.x == 0) fgw[b] = fmaxf(s, 1.f);
}

// ---------------- kernel 1: build pixel tables ----------------
// packed[(bv*NTILE+tile)*32 + lane] = per-lane float4 (bdot.x,bdot.y,bq.x,bq.y)
// for the 16x16x4 f32 B fragment (K rows striped: VGPR0 = K0/K1 halves, VGPR1 = K2/K3).
__global__ __launch_bounds__(256) void build_tables(const int* __restrict__ mask,
                                                    const float* __restrict__ vp,
                                                    float* __restrict__ packed,
                                                    float* __restrict__ Ptab) {
  int gid = blockIdx.x*256 + threadIdx.x;
  if (gid >= Bn*VNv*TNp) return;
  int t  = gid % TNp;
  int bv = gid / TNp;
  int v  = bv % VNv;
  int b  = bv / VNv;
  float ux = vp[(size_t)(b*2*VNv + 2*v)*TNp + t];
  float uy = vp[(size_t)(b*2*VNv + 2*v + 1)*TNp + t];
  float px = (float)(t % Wn), py = (float)(t / Wn);
  float cc = px*ux + py*uy;                 // c = p . u
  float uu = ux*ux + uy*uy;
  float uq = (mask[b*TNp + t] > 0) ? (THR2*uu) : BIGQ;

  ((float4*)Ptab)[(size_t)bv*TNp + t] = make_float4(ux, uy, cc, uq);

  int tile = t >> 4, n = t & 15;
  size_t base = ((size_t)bv*NTILE + tile)*32;
  // lane n (half 0): (Kdot0, Kdot2, Kq0, Kq2)
  ((float4*)packed)[base + n]      = make_float4(ux, cc, uq, uq*py);
  // lane n+16 (half 1): (Kdot1, Kdot3, Kq1, Kq3)
  ((float4*)packed)[base + n + 16] = make_float4(uy, 0.f, uq*px, uq*(px*px + py*py));
}

// ---------------- kernel 2: WMMA voting ----------------
// One block = 256 threads = 8 waves; each wave owns NJ 16-hyp tiles.
// dot(h,t) and q(h,t) are K=4 bilinear forms -> V_WMMA_F32_16X16X4_F32.
// LDS pixel tiles are double-buffered: chunk c+1 streams in via async DMA
// (ASYNCcnt) while chunk c feeds the WMMA pipe from the other buffer.
template<int NJ>
__global__ __launch_bounds__(256) void vote_kernel(const float* __restrict__ vp,
                                                   const int* __restrict__ idxs,
                                                   const float* __restrict__ packed,
                                                   float* __restrict__ cnt_out,
                                                   float* __restrict__ hyp_out,
                                                   int nhyp) {
  constexpr int HPB = NJ * 128;             // hyps per block (8 waves * NJ tiles * 16)
  constexpr int LPT = (CHT*32)/256;         // async b128 loads per thread per chunk (12)
  __shared__ float  s_hx[512], s_hy[512];
  __shared__ float4 s_buf[2*CHT*32];        // 2 x 48KB staging buffers

  int S  = nhyp / HPB;
  int bv = blockIdx.x / S;
  int s  = blockIdx.x % S;
  int b  = bv / VNv, v = bv % VNv;
  int h0 = s * HPB;

  // Phase A: generate this block's hypotheses
  for (int i = threadIdx.x; i < HPB; i += 256) {
    int h = h0 + i;
    const int* ip = idxs + ((size_t)(b*nhyp + h)*VNv + v)*2;
    float2 hp = gen_hyp(vp, ip, b, v);
    s_hx[i] = hp.x; s_hy[i] = hp.y;
    hyp_out[((size_t)bv*nhyp + h)*2 + 0] = hp.x;
    hyp_out[((size_t)bv*nhyp + h)*2 + 1] = hp.y;
  }
  __syncthreads();

  int wave = threadIdx.x >> 5, lane = threadIdx.x & 31;

  // A fragments (16x4 f32: lanes 0-15 hold K0/K1 of row M, lanes 16-31 hold K2/K3)
  v2f adot[NJ], aq[NJ];
  float acc[NJ][8];
#pragma unroll
  for (int j = 0; j < NJ; ++j) {
    int m = (wave + 8*j)*16 + (lane & 15);
    float hx = s_hx[m], hy = s_hy[m];
    if (lane < 16) {
      adot[j].x = hx;            adot[j].y = hy;
      aq[j].x   = hx*hx + hy*hy; aq[j].y   = -2.f*hx;
    } else {
      adot[j].x = -1.f;          adot[j].y = 0.f;
      aq[j].x   = -2.f*hy;       aq[j].y   = 1.f;
    }
#pragma unroll
    for (int e = 0; e < 8; ++e) acc[j][e] = 0.f;
  }

  const float4* pk = (const float4*)packed + (size_t)bv*NTILE*32;
  uint32_t lbase = (uint32_t)(uintptr_t)(&s_buf[0]);

  // vote step for one per-lane B fragment
  auto process = [&](float4 pv) {
    v2f bdot; bdot.x = pv.x; bdot.y = pv.y;
    v2f bq;   bq.x   = pv.z; bq.y   = pv.w;
#pragma unroll
    for (int j = 0; j < NJ; ++j) {
      v8f z = {};
      v8f dd = __builtin_amdgcn_wmma_f32_16x16x4_f32(false, adot[j], false, bdot,
                                                     (short)0, z, false, false);
      v8f qq = __builtin_amdgcn_wmma_f32_16x16x4_f32(false, aq[j],  false, bq,
                                                     (short)0, z, false, false);
#pragma unroll
      for (int e = 0; e < 8; ++e) {
        float d = dd[e];
        acc[j][e] += ((d > 0.f) && (d*d > qq[e])) ? 1.f : 0.f;
      }
    }
  };

  // prime: stream chunk 0 into buffer 0
  {
    const float4* src = pk;
#pragma unroll
    for (int k = 0; k < LPT; ++k) {
      int i = threadIdx.x + k*256;
      async_ld_b128(lbase + (uint32_t)(i*16), (const void*)(src + i));
    }
  }

  for (int c = 0; c < NCHUNK; ++c) {
    // prefetch chunk c+1 into the other buffer while we compute on chunk c
    if (c + 1 < NCHUNK) {
      const float4* src = pk + (size_t)(c + 1)*CHT*32;
      uint32_t dst = lbase + (uint32_t)(((c + 1) & 1)*CHT*32*16);
#pragma unroll
      for (int k = 0; k < LPT; ++k) {
        int i = threadIdx.x + k*256;
        async_ld_b128(dst + (uint32_t)(i*16), (const void*)(src + i));
      }
      wait_async_keep12();     // chunk c landed; chunk c+1 still in flight
    } else {
      wait_async0();           // last chunk: drain everything
    }
    __syncthreads();           // buffer handoff: all waves' DMA for chunk c done

    const float4* bufc = s_buf + (size_t)(c & 1)*CHT*32;
    // software-pipelined LDS reads: fetch tt+1's fragment before using tt's,
    // so the ds_load latency is hidden behind the WMMA+vote work.
    float4 pv = bufc[lane];
    for (int tt = 0; tt < CHT - 1; ++tt) {
      float4 nv = bufc[(tt + 1)*32 + lane];
      process(pv);
      pv = nv;
    }
    process(pv);

    __syncthreads();           // all waves done reading buf (c&1) before reuse
  }

  // Phase C: reduce pixel-lane partials (D: VGPR e holds M=e / M=e+8 per half)
#pragma unroll
  for (int j = 0; j < NJ; ++j) {
#pragma unroll
    for (int e = 0; e < 8; ++e) {
      float vs = acc[j][e];
      vs += __shfl_xor(vs, 1, 32);
      vs += __shfl_xor(vs, 2, 32);
      vs += __shfl_xor(vs, 4, 32);
      vs += __shfl_xor(vs, 8, 32);
      acc[j][e] = vs;
    }
    int hbase = h0 + (wave + 8*j)*16;
    if (lane == 0) {
#pragma unroll
      for (int e = 0; e < 8; ++e) cnt_out[(size_t)bv*nhyp + hbase + e] = acc[j][e];
    } else if (lane == 16) {
#pragma unroll
      for (int e = 0; e < 8; ++e) cnt_out[(size_t)bv*nhyp + hbase + 8 + e] = acc[j][e];
    }
  }
}

// ---------------- kernel 3: RANSAC argmax + weighted least squares ----------------
__global__ __launch_bounds__(256) void ransac_finish(const float* __restrict__ cnt,
                                                     const float* __restrict__ hypw,
                                                     const float* __restrict__ Ptab,
                                                     float* __restrict__ meanw,
                                                     float* __restrict__ out) {
  __shared__ float s_val[256];
  __shared__ int   s_idx[256];
  __shared__ float s8[8];
  int bv = blockIdx.x;

  float best = -1.f; int bi = RH;
  for (int h = threadIdx.x; h < RH; h += 256) {
    float c = cnt[(size_t)bv*RH + h];
    if (c > best) { best = c; bi = h; }
  }
  s_val[threadIdx.x] = best; s_idx[threadIdx.x] = bi;
  __syncthreads();
  for (int st = 128; st > 0; st >>= 1) {
    if (threadIdx.x < st) {
      float ov = s_val[threadIdx.x + st]; int oi = s_idx[threadIdx.x + st];
      if (ov > s_val[threadIdx.x] ||
          (ov == s_val[threadIdx.x] && oi < s_idx[threadIdx.x])) {
        s_val[threadIdx.x] = ov; s_idx[threadIdx.x] = oi;
      }
    }
    __syncthreads();
  }
  int win = s_idx[0];
  float wx = hypw[((size_t)bv*RH + win)*2 + 0];
  float wy = hypw[((size_t)bv*RH + win)*2 + 1];

  const float4* P = (const float4*)Ptab + (size_t)bv*TNp;
  float s00 = 0, s01 = 0, s11 = 0, r0 = 0, r1 = 0;
  for (int t = threadIdx.x; t < TNp; t += 256) {
    float4 p = P[t];                                   // (ux,uy,c,uq)
    float px = (float)(t % Wn), py = (float)(t / Wn);
    float dot = wx*p.x + wy*p.y - p.z;
    float dxp = wx - px, dyp = wy - py;
    float dd = dxp*dxp + dyp*dyp;
    if ((dot > 0.f) && (dot*dot > p.w*dd)) {
      float nx = p.y, ny = -p.x;                       // n = (uy, -ux)
      float bt = nx*px + ny*py;
      s00 += nx*nx; s01 += nx*ny; s11 += ny*ny;
      r0  += nx*bt; r1  += ny*bt;
    }
  }
  s00 = block_sum(s00, s8);
  s01 = block_sum(s01, s8);
  s11 = block_sum(s11, s8);
  r0  = block_sum(r0,  s8);
  r1  = block_sum(r1,  s8);
  if (threadIdx.x == 0) {
    float a00 = s00 + EPSF, a11 = s11 + EPSF, a01 = s01;
    float inv = 1.f / (a00*a11 - a01*a01);
    float mx = (a11*r0 - a01*r1) * inv;
    float my = (a00*r1 - a01*r0) * inv;
    meanw[bv*2 + 0] = mx; meanw[bv*2 + 1] = my;
    out[bv*2 + 0]   = mx; out[bv*2 + 1]   = my;
  }
}

// ---------------- kernel 4: thresholded weighted covariance ----------------
__global__ __launch_bounds__(256) void cov_kernel(const float* __restrict__ cnt,
                                                  const float* __restrict__ hypw,
                                                  const float* __restrict__ meanw,
                                                  const float* __restrict__ fgw,
                                                  float* __restrict__ out2) {
  __shared__ float s8[8];
  int bv = blockIdx.x;
  int b  = bv / VNv;
  float inv_fg = 1.f / fgw[b];

  float mx = -1.f;
  for (int h = threadIdx.x; h < DH; h += 256)
    mx = fmaxf(mx, cnt[(size_t)bv*DH + h] * inv_fg);
  float thr = block_max(mx, s8) - 0.1f;

  float mex = meanw[bv*2 + 0], mey = meanw[bv*2 + 1];
  float c00 = 0, c01 = 0, c11 = 0, sr = 0;
  for (int h = threadIdx.x; h < DH; h += 256) {
    float r = cnt[(size_t)bv*DH + h] * inv_fg;
    r = (r >= thr) ? r : 0.f;
    float dx = hypw[((size_t)bv*DH + h)*2 + 0] - mex;
    float dy = hypw[((size_t)bv*DH + h)*2 + 1] - mey;
    c00 += dx*dx*r; c01 += dx*dy*r; c11 += dy*dy*r; sr += r;
  }
  c00 = block_sum(c00, s8);
  c01 = block_sum(c01, s8);
  c11 = block_sum(c11, s8);
  sr  = block_sum(sr,  s8);
  if (threadIdx.x == 0) {
    float inv = 1.f / fmaxf(sr, 0.001f);
    out2[bv*4 + 0] = c00*inv;
    out2[bv*4 + 1] = c01*inv;
    out2[bv*4 + 2] = c01*inv;
    out2[bv*4 + 3] = c11*inv;
  }
}

// ---------------- launch ----------------
extern "C" void kernel_launch(void* const* d_in, const int* in_sizes, int n_in,
                              void* d_out, int out_size, void* d_ws, size_t ws_size,
                              hipStream_t stream) {
  (void)in_sizes; (void)n_in; (void)out_size; (void)ws_size;
  const int*   mask = (const int*)d_in[0];
  const float* vp   = (const float*)d_in[1];
  const int*   ridx = (const int*)d_in[2];
  const int*   didx = (const int*)d_in[3];
  float* out = (float*)d_out;
  float* ws  = (float*)d_ws;

  float* packed = ws + OFF_PACKED;
  float* Ptab   = ws + OFF_P;
  float* fgw    = ws + OFF_FG;
  float* meanw  = ws + OFF_MEAN;
  float* cntr   = ws + OFF_CNTR;
  float* hypr   = ws + OFF_HYPR;
  float* cntd   = ws + OFF_CNTD;
  float* hypd   = ws + OFF_HYPD;

  fg_kernel<<<Bn, 256, 0, stream>>>(mask, fgw);
  build_tables<<<(Bn*VNv*TNp + 255)/256, 256, 0, stream>>>(mask, vp, packed, Ptab);
  // RANSAC: 128 hyps/block -> 4 splits per (b,v) -> 72 blocks
  vote_kernel<1><<<Bn*VNv*(RH/128), 256, 0, stream>>>(vp, ridx, packed, cntr, hypr, RH);
  ransac_finish<<<Bn*VNv, 256, 0, stream>>>(cntr, hypr, Ptab, meanw, out);
  // DIST: 512 hyps/block -> 8 splits per (b,v) -> 144 blocks
  vote_kernel<4><<<Bn*VNv*(DH/512), 256, 0, stream>>>(vp, didx, packed, cntd, hypd, DH);
  cov_kernel<<<Bn*VNv, 256, 0, stream>>>(cntd, hypd, meanw, fgw, out + Bn*VNv*2);
}